// GAPLayer_single_48945447305280
// MI455X (gfx1250) — compile-verified
//
#include <hip/hip_runtime.h>

// Problem constants (from reference)
constexpr int BB  = 8;
constexpr int NN  = 32768;
constexpr int KK  = 20;
constexpr int CH  = 16;
constexpr int PPW = 4;                      // points per wave
constexpr int WPB = 8;                      // waves per block (256 threads)
constexpr float EPS = 1e-5f;

typedef __attribute__((ext_vector_type(2))) float v2f;
typedef __attribute__((ext_vector_type(8))) float v8f;

// ds_swizzle group-of-32 xor pattern: offset[14:10]=xor_mask, [9:5]=or, [4:0]=and
#define SWZ_XOR(m) (((m) << 10) | 0x1F)

template <int PATT>
__device__ __forceinline__ float swz(float v) {
    return __int_as_float(__builtin_amdgcn_ds_swizzle(__float_as_int(v), PATT));
}

// lrelu(t) = max(t, 0.2t); fmaxf form dual-issues well on gfx1250
__device__ __forceinline__ float lrelu(float t) { return fmaxf(t, 0.2f * t); }

// sum across the 16 lanes of each half-wave; result replicated in all 16 lanes
__device__ __forceinline__ float dot16(float v) {
    v += swz<SWZ_XOR(1)>(v);
    v += swz<SWZ_XOR(2)>(v);
    v += swz<SWZ_XOR(4)>(v);
    v += swz<SWZ_XOR(8)>(v);
    return v;
}

__global__ __launch_bounds__(32 * WPB) void gap_wmma_kernel(
    const float* __restrict__ x,            // (B,1,N)
    const int*   __restrict__ idx32,        // (B,N,K) int64 viewed as int32 pairs (LE low dword)
    const float* __restrict__ w1,  const float* __restrict__ g1,  const float* __restrict__ b1,
    const float* __restrict__ m1,  const float* __restrict__ v1,
    const float* __restrict__ w2,  const float* __restrict__ g2,  const float* __restrict__ b2,
    const float* __restrict__ m2,  const float* __restrict__ v2,
    const float* __restrict__ w1n, const float* __restrict__ g1n, const float* __restrict__ b1n,
    const float* __restrict__ m1n, const float* __restrict__ v1n,
    const float* __restrict__ w2n, const float* __restrict__ g2n, const float* __restrict__ b2n,
    const float* __restrict__ m2n, const float* __restrict__ v2n,
    float* __restrict__ out,                // (B,N,CH)
    float* __restrict__ edge_o)             // (B,N,K,CH)
{
    const int lane = threadIdx.x & 31;
    // Force the wave id into an SGPR so all per-point bases/addresses are scalar:
    // gathers become saddr+voffset (scale_offset), xc becomes a scalar load, and
    // the streaming stores become SADDR + lane-constant offset + immediate.
    const int wave = __builtin_amdgcn_readfirstlane(threadIdx.x >> 5);
    const int l16  = lane & 15;
    const bool hi  = lane >= 16;
    const int o    = l16;                   // output channel owned by this lane

    // ---- Fold BN into weights (once per wave, per-lane channel o) ----
    float s1 = g1[o] * rsqrtf(v1[o] + EPS);
    v2f B1;
    {
        float wa = w1[2*o] * s1, wb = w1[2*o+1] * s1;
        float c1 = b1[o] - m1[o] * s1;
        B1[0] = hi ? c1 : wa;               // hi half-lanes carry K2 = bias row
        B1[1] = hi ? 0.f : wb;              // K3 = 0
    }
    float s1n = g1n[o] * rsqrtf(v1n[o] + EPS);
    v2f B1N;
    {
        float wa = w1n[2*o] * s1n, wb = w1n[2*o+1] * s1n;
        float c1n = b1n[o] - m1n[o] * s1n;
        B1N[0] = hi ? c1n : wa;
        B1N[1] = hi ? 0.f : wb;
    }
    float s2  = g2[0]  * rsqrtf(v2[0]  + EPS);
    float s2n = g2n[0] * rsqrtf(v2n[0] + EPS);
    float w2p  = w2[o]  * s2;
    float w2np = w2n[o] * s2n;
    float cbias = (b2[0] - m2[0]*s2) + (b2n[0] - m2n[0]*s2n);

    // Grid exactly covers B*N points: no bounds checks needed.
    const int p0 = (blockIdx.x * WPB + wave) * PPW;

    // Per-lane edge id (constant across points): lo half -> k=l16 (tile0 rows),
    // hi half -> k=16+l16 (tile1 rows 0..3). Clamp OOB, mask f.
    const int  myk = hi ? (16 + l16) : l16;
    const bool kok = myk < KK;
    const int  kc  = kok ? myk : 0;

    for (int pi = 0; pi < PPW; ++pi) {
        const int bn = p0 + pi;             // scalar flat (b,n); b = bn>>15, n = bn&32767
        const float* xb = x + ((long long)(bn >> 15) << 15);   // scalar: x + b*NN

        const float xc = xb[bn & (NN - 1)];                    // scalar load

        // 32-bit gather: low dword of int64 idx (values < 32768); scalar base + lane offset
        int j = idx32[bn * (2 * KK) + 2 * kc];
        float xj = xb[j];                                      // saddr + scaled voffset
        float f  = kok ? (xj - xc) : 0.f;

        // Tile0 A (16x4 f32): row m: [f_m, xc, 1, 0]
        v2f A0, A1;
        A0[0] = hi ? 1.f : f;
        A0[1] = hi ? 0.f : xc;
        // Tile1: rows 0..3 valid (k=16..19); f comes from the hi half via xor-16 swizzle
        float f1 = swz<SWZ_XOR(16)>(f);
        bool row1 = (!hi) && (l16 < 4);
        A1[0] = hi ? 1.f : (row1 ? f1 : 0.f);
        A1[1] = hi ? 0.f : (row1 ? xc : 0.f);

        v8f z = {};
        v8f h0 = __builtin_amdgcn_wmma_f32_16x16x4_f32(false, A0, false, B1,  (short)0, z, false, false);
        v8f e0 = __builtin_amdgcn_wmma_f32_16x16x4_f32(false, A0, false, B1N, (short)0, z, false, false);
        v8f h1 = __builtin_amdgcn_wmma_f32_16x16x4_f32(false, A1, false, B1,  (short)0, z, false, false);
        v8f e1 = __builtin_amdgcn_wmma_f32_16x16x4_f32(false, A1, false, B1N, (short)0, z, false, false);

        // LReLU + fused attention-logit dot (a+e branches combined => one dot16 per edge)
        float ap0[8], ap1[4], e1v[4];
        #pragma unroll
        for (int r = 0; r < 8; ++r) {
            float hh = lrelu(h0[r]);
            float ee = lrelu(e0[r]);
            e0[r] = ee;                                   // edge_o values (tile0)
            float d = dot16(__builtin_fmaf(hh, w2p, ee * w2np)) + cbias;
            ap0[r] = lrelu(d);
        }
        #pragma unroll
        for (int r = 0; r < 4; ++r) {                     // only rows 0..3 of tile1 are live
            float hh = lrelu(h1[r]);
            float ee = lrelu(e1[r]);
            e1v[r] = ee;                                  // edge_o values (tile1 lo-half rows)
            float d = dot16(__builtin_fmaf(hh, w2p, ee * w2np)) + cbias;
            ap1[r] = lrelu(d);
        }

        // Softmax over k=0..19.
        // tile0 component r -> edge k = hi ? 8+r : r (all valid)
        // tile1 component r -> edge k = 16+r, valid only on lo half with r<4
        const float NEG = -3.0e38f;
        float mx = NEG;
        #pragma unroll
        for (int r = 0; r < 8; ++r) mx = fmaxf(mx, ap0[r]);
        #pragma unroll
        for (int r = 0; r < 4; ++r) mx = fmaxf(mx, hi ? NEG : ap1[r]);
        mx = fmaxf(mx, swz<SWZ_XOR(16)>(mx));             // full max over 20 edges, all lanes

        float se = 0.f;
        float at0[8], at1[4];
        #pragma unroll
        for (int r = 0; r < 8; ++r) { at0[r] = __expf(ap0[r] - mx); se += at0[r]; }
        #pragma unroll
        for (int r = 0; r < 4; ++r) { at1[r] = hi ? 0.f : __expf(ap1[r] - mx); se += at1[r]; }
        se += swz<SWZ_XOR(16)>(se);                       // full sum, all lanes
        float inv = 1.0f / se;

        // out[b][n][o] = sum_k att[k] * edge_o[k][o]  (edges are in-lane components here)
        float po = 0.f;
        #pragma unroll
        for (int r = 0; r < 8; ++r) po = __builtin_fmaf(at0[r], e0[r], po);
        #pragma unroll
        for (int r = 0; r < 4; ++r) po = __builtin_fmaf(at1[r], e1v[r], po);
        po = (po + swz<SWZ_XOR(16)>(po)) * inv;
        if (!hi) __builtin_nontemporal_store(po, &out[(long long)bn * CH + o]);

        // Store edge_o (streaming 335 MB -> non-temporal; scalar base per point,
        // lane offset o*4, immediate k*64; 32 lanes per component cover two
        // contiguous 64B segments)
        float* eob = edge_o + (long long)bn * (KK * CH) + o;
        #pragma unroll
        for (int r = 0; r < 8; ++r) {
            int k = hi ? (8 + r) : r;
            __builtin_nontemporal_store(e0[r], &eob[k * CH]);
        }
        if (!hi) {
            #pragma unroll
            for (int r = 0; r < 4; ++r)
                __builtin_nontemporal_store(e1v[r], &eob[(16 + r) * CH]);
        }
    }
}

extern "C" void kernel_launch(void* const* d_in, const int* in_sizes, int n_in,
                              void* d_out, int out_size, void* d_ws, size_t ws_size,
                              hipStream_t stream) {
    const float* x     = (const float*)d_in[0];
    // d_in[1] = pos (unused), d_in[3] = dis (unused)
    const int*   idx32 = (const int*)d_in[2];   // int64 viewed as LE int32 pairs
    const float* w1  = (const float*)d_in[4];
    const float* g1  = (const float*)d_in[5];
    const float* b1  = (const float*)d_in[6];
    const float* m1  = (const float*)d_in[7];
    const float* v1  = (const float*)d_in[8];
    const float* w2  = (const float*)d_in[9];
    const float* g2  = (const float*)d_in[10];
    const float* b2  = (const float*)d_in[11];
    const float* m2  = (const float*)d_in[12];
    const float* v2  = (const float*)d_in[13];
    const float* w1n = (const float*)d_in[14];
    const float* g1n = (const float*)d_in[15];
    const float* b1n = (const float*)d_in[16];
    const float* m1n = (const float*)d_in[17];
    const float* v1n = (const float*)d_in[18];
    const float* w2n = (const float*)d_in[19];
    const float* g2n = (const float*)d_in[20];
    const float* b2n = (const float*)d_in[21];
    const float* m2n = (const float*)d_in[22];
    const float* v2n = (const float*)d_in[23];

    float* out    = (float*)d_out;
    float* edge_o = out + (size_t)BB * NN * CH;

    const long long P = (long long)BB * NN;          // 262144, divisible by PPW*WPB
    int blocks = (int)(P / (PPW * WPB));             // 8192

    gap_wmma_kernel<<<blocks, 32 * WPB, 0, stream>>>(
        x, idx32,
        w1, g1, b1, m1, v1, w2, g2, b2, m2, v2,
        w1n, g1n, b1n, m1n, v1n, w2n, g2n, b2n, m2n, v2n,
        out, edge_o);
}